// weight_nd_Loss_54065048323036
// MI455X (gfx1250) — compile-verified
//
#include <hip/hip_runtime.h>

#define BATCH 4096
#define DFEAT 128
#define NCLS  512
#define NEGINF (-1e30f)
#define BT   64   // tile edge (M and N)
#define KC   64   // K chunk staged in LDS
#define KPAD 68   // padded LDS row stride (floats): +4 skews banks

typedef float v2f __attribute__((ext_vector_type(2)));
typedef float v8f __attribute__((ext_vector_type(8)));

__device__ __forceinline__ void online_upd(float& m, float& s1, float& s2, float l, float d) {
  float nm = fmaxf(m, l);
  float r  = __expf(m - nm);
  float e  = __expf(l - nm);
  s1 = s1 * r + e;
  s2 = s2 * r + e * d;
  m  = nm;
}

__device__ __forceinline__ void combine_lse(float& m, float& s1, float& s2,
                                            float mo, float s1o, float s2o) {
  float nm = fmaxf(m, mo);
  float ra = __expf(m - nm);
  float rb = __expf(mo - nm);
  s1 = s1 * ra + s1o * rb;
  s2 = s2 * ra + s2o * rb;
  m  = nm;
}

// ---------------- kernel 0: row squared norms -------------------------------
__global__ __launch_bounds__(256)
void sq_kernel(const float* __restrict__ x, float* __restrict__ sq) {
  int i = blockIdx.x * 256 + threadIdx.x;
  if (i >= BATCH) return;
  const float4* row = (const float4*)(x + (size_t)i * DFEAT);
  float s = 0.f;
#pragma unroll 8
  for (int k = 0; k < DFEAT / 4; ++k) {
    float4 v = row[k];
    s += v.x * v.x + v.y * v.y + v.z * v.z + v.w * v.w;
  }
  sq[i] = s;
}

// ---------------- kernel 1: fused WMMA dist tile + loss partials ------------
// 128 threads = 4 wave32; each wave owns a 16x64 strip of the 64x64 tile.
__global__ __launch_bounds__(128)
void dist_loss_tile_kernel(const float* __restrict__ x, const int* __restrict__ y,
                           const float* __restrict__ meanm, const float* __restrict__ stdm,
                           const float* __restrict__ noise, const float* __restrict__ sq,
                           float* __restrict__ dist, float* __restrict__ partials)
{
  __shared__ float ldsA[BT][KPAD];
  __shared__ float ldsB[BT][KPAD];
  __shared__ float red[4][12];

  const int tileM = blockIdx.y * BT;
  const int tileN = blockIdx.x * BT;
  const int tid   = threadIdx.x;
  const int lane  = tid & 31;
  const int wave  = tid >> 5;
  const int lrow  = lane & 15;     // M (for A) / N (for B) index within 16
  const int khalf = lane >> 4;     // which K pair this half-wave holds

  v8f acc[4] = {};

  for (int kc = 0; kc < DFEAT; kc += KC) {
    __syncthreads();
    // cooperative stage: 64 rows x 64 cols of both row-tiles, float4 loads
    for (int idx = tid; idx < BT * (KC / 4); idx += 128) {
      int r  = idx >> 4;
      int c4 = idx & 15;
      float4 va = *(const float4*)(x + (size_t)(tileM + r) * DFEAT + kc + c4 * 4);
      float4 vb = *(const float4*)(x + (size_t)(tileN + r) * DFEAT + kc + c4 * 4);
      ldsA[r][c4 * 4 + 0] = va.x; ldsA[r][c4 * 4 + 1] = va.y;
      ldsA[r][c4 * 4 + 2] = va.z; ldsA[r][c4 * 4 + 3] = va.w;
      ldsB[r][c4 * 4 + 0] = vb.x; ldsB[r][c4 * 4 + 1] = vb.y;
      ldsB[r][c4 * 4 + 2] = vb.z; ldsB[r][c4 * 4 + 3] = vb.w;
    }
    __syncthreads();

    const int mrow = wave * 16 + lrow;
    for (int k0 = 0; k0 < KC; k0 += 4) {
      const int kk = k0 + 2 * khalf;              // even -> 8B-aligned ds_load_b64
      v2f a = *(const v2f*)&ldsA[mrow][kk];
#pragma unroll
      for (int t = 0; t < 4; ++t) {
        v2f b = *(const v2f*)&ldsB[t * 16 + lrow][kk];
        acc[t] = __builtin_amdgcn_wmma_f32_16x16x4_f32(
            false, a, false, b, (short)0, acc[t], false, false);
      }
    }
  }

  // ---------------- epilogue: dist write + masked-softmax partials ----------
  float sqi[8];
  int   yi[8];
  const int ibase = tileM + wave * 16 + 8 * khalf;   // C layout: M = v + 8*(lane/16)
#pragma unroll
  for (int v = 0; v < 8; ++v) {
    sqi[v] = sq[ibase + v];
    yi[v]  = y[ibase + v];
  }

  float m1 = NEGINF, s11 = 0.f, s12 = 0.f;
  float m2 = NEGINF, s21 = 0.f, s22 = 0.f;
  float m3 = NEGINF, s31 = 0.f, s32 = 0.f;
  float pf = 0.f, pf2 = 0.f, pc = 0.f;

#pragma unroll
  for (int t = 0; t < 4; ++t) {
    const int   j   = tileN + t * 16 + lrow;
    const float sqj = sq[j];
    const int   yj  = y[j];
#pragma unroll
    for (int v = 0; v < 8; ++v) {
      const int   i = ibase + v;
      const float d = -2.0f * acc[t][v] + sqi[v] + sqj;
      // dist is written exactly once and never re-read: stream it (TH=NT)
      __builtin_nontemporal_store(d, &dist[(size_t)i * BATCH + j]);
      if (j > i) {                                  // strict upper triangle
        const size_t ci  = (size_t)yi[v] * NCLS + yj;
        const float  sd  = stdm[ci];
        // noise is read exactly once: non-temporal, keep L2 for x/mean/std
        const float  nz  = __builtin_nontemporal_load(&noise[(size_t)i * BATCH + j]);
        const float  obs = meanm[ci] + sd * nz;
        const bool same = ((i >> 3) == (j >> 3));   // label = index/INSTANCES
        if (!same) {
          const float tn1 = obs - 2.5f * sd;
          const float tmd = obs - 1.5f * sd;
          const float thi = obs + 1.5f * sd;
          if (d < tn1)                      online_upd(m1, s11, s12, 10.f * (tn1 - d), d);
          else if (d > tn1 && d < tmd)      online_upd(m2, s21, s22,  5.f * (tmd - d), d);
          else if (d > tmd && d < thi)      online_upd(m3, s31, s32,  5.f * (thi - d), d);
        } else {
          const float tp = obs + 2.0f * sd;
          if (d > tp) { pf += d; pf2 += d * d; pc += 1.f; }
        }
      }
    }
  }

  // wave32 tree reduce of the 12-value record
  float vals[12] = {m1, s11, s12, m2, s21, s22, m3, s31, s32, pf, pf2, pc};
#pragma unroll
  for (int off = 16; off > 0; off >>= 1) {
    float o[12];
#pragma unroll
    for (int q = 0; q < 12; ++q) o[q] = __shfl_xor(vals[q], off, 32);
    combine_lse(vals[0], vals[1], vals[2], o[0], o[1], o[2]);
    combine_lse(vals[3], vals[4], vals[5], o[3], o[4], o[5]);
    combine_lse(vals[6], vals[7], vals[8], o[6], o[7], o[8]);
    vals[9] += o[9]; vals[10] += o[10]; vals[11] += o[11];
  }
  if (lane == 0) {
#pragma unroll
    for (int q = 0; q < 12; ++q) red[wave][q] = vals[q];
  }
  __syncthreads();
  if (tid == 0) {
    float g[12];
#pragma unroll
    for (int q = 0; q < 12; ++q) g[q] = red[0][q];
    for (int w = 1; w < 4; ++w) {
      combine_lse(g[0], g[1], g[2], red[w][0], red[w][1], red[w][2]);
      combine_lse(g[3], g[4], g[5], red[w][3], red[w][4], red[w][5]);
      combine_lse(g[6], g[7], g[8], red[w][6], red[w][7], red[w][8]);
      g[9] += red[w][9]; g[10] += red[w][10]; g[11] += red[w][11];
    }
    float* p = partials + (size_t)(blockIdx.y * gridDim.x + blockIdx.x) * 12;
#pragma unroll
    for (int q = 0; q < 12; ++q) p[q] = g[q];
  }
}

// ---------------- kernel 2: fold block records into the scalar loss ---------
__global__ __launch_bounds__(256)
void final_reduce_kernel(const float* __restrict__ part, int nrec,
                         float* __restrict__ loss_out)
{
  __shared__ float red[8][12];
  float v[12] = {NEGINF, 0, 0, NEGINF, 0, 0, NEGINF, 0, 0, 0, 0, 0};
  for (int r = threadIdx.x; r < nrec; r += 256) {
    const float* p = part + (size_t)r * 12;
    combine_lse(v[0], v[1], v[2], p[0], p[1], p[2]);
    combine_lse(v[3], v[4], v[5], p[3], p[4], p[5]);
    combine_lse(v[6], v[7], v[8], p[6], p[7], p[8]);
    v[9] += p[9]; v[10] += p[10]; v[11] += p[11];
  }
#pragma unroll
  for (int off = 16; off > 0; off >>= 1) {
    float o[12];
#pragma unroll
    for (int q = 0; q < 12; ++q) o[q] = __shfl_xor(v[q], off, 32);
    combine_lse(v[0], v[1], v[2], o[0], o[1], o[2]);
    combine_lse(v[3], v[4], v[5], o[3], o[4], o[5]);
    combine_lse(v[6], v[7], v[8], o[6], o[7], o[8]);
    v[9] += o[9]; v[10] += o[10]; v[11] += o[11];
  }
  const int lane = threadIdx.x & 31, wave = threadIdx.x >> 5;
  if (lane == 0) {
#pragma unroll
    for (int q = 0; q < 12; ++q) red[wave][q] = v[q];
  }
  __syncthreads();
  if (threadIdx.x == 0) {
    float g[12];
#pragma unroll
    for (int q = 0; q < 12; ++q) g[q] = red[0][q];
    for (int w = 1; w < 8; ++w) {
      combine_lse(g[0], g[1], g[2], red[w][0], red[w][1], red[w][2]);
      combine_lse(g[3], g[4], g[5], red[w][3], red[w][4], red[w][5]);
      combine_lse(g[6], g[7], g[8], red[w][6], red[w][7], red[w][8]);
      g[9] += red[w][9]; g[10] += red[w][10]; g[11] += red[w][11];
    }
    const float ln1 = (g[1]  > 0.f) ? (-g[2] / g[1]) : -0.0001f;
    const float ln2 = (g[4]  > 0.f) ? (-g[5] / g[4]) : -0.0001f;
    const float ln3 = (g[7]  > 0.f) ? (-g[8] / g[7]) : -0.0001f;
    const float lp  = (g[11] > 0.f) ? (g[10] / ((g[9] != 0.f) ? g[9] : 1.f)) : 0.0001f;
    loss_out[0] = 1.0f + 0.25f * (ln1 + ln2 + ln3 + lp);
  }
}

// ---------------- launch ----------------------------------------------------
extern "C" void kernel_launch(void* const* d_in, const int* in_sizes, int n_in,
                              void* d_out, int out_size, void* d_ws, size_t ws_size,
                              hipStream_t stream) {
  (void)in_sizes; (void)n_in; (void)out_size; (void)ws_size;
  const float* x     = (const float*)d_in[0];
  const int*   y     = (const int*)d_in[1];
  const float* meanm = (const float*)d_in[2];
  const float* stdm  = (const float*)d_in[3];
  const float* noise = (const float*)d_in[4];
  float* out = (float*)d_out;

  float* sq       = (float*)d_ws;            // BATCH floats
  float* partials = sq + BATCH;              // (BATCH/BT)^2 * 12 floats

  sq_kernel<<<BATCH / 256, 256, 0, stream>>>(x, sq);

  dim3 grid(BATCH / BT, BATCH / BT);
  dist_loss_tile_kernel<<<grid, 128, 0, stream>>>(x, y, meanm, stdm, noise, sq,
                                                  out, partials);

  const int nrec = (BATCH / BT) * (BATCH / BT);
  final_reduce_kernel<<<1, 256, 0, stream>>>(partials, nrec,
                                             out + (size_t)BATCH * BATCH);
}